// SlotPrompt_68539088109656
// MI455X (gfx1250) — compile-verified
//
#include <hip/hip_runtime.h>

#define BSZ   256
#define KSLOT 8
#define H     128
#define P     16384
#define NFLAT (BSZ * KSLOT)   // 2048 flat slots
#define ALPHA 0.8f

typedef __attribute__((ext_vector_type(2))) float v2f;
typedef __attribute__((ext_vector_type(8))) float v8f;

// ---------------------------------------------------------------------------
// Kernel 1: batched nearest-neighbor (match_pool) via fp32 WMMA.
// One wave (32 lanes) owns a 16-query block; sweeps the pool in 16-col tiles.
// Distance tile: D = ||p||^2 - 2 * q.p   (||q||^2 dropped: row-constant).
// q.p computed with 32 chained V_WMMA_F32_16X16X4_F32 (K=128 = 32*4).
// ---------------------------------------------------------------------------
__global__ __launch_bounds__(256)
void slotprompt_match_kernel(const float* __restrict__ slots,
                             const float* __restrict__ pool,
                             float* __restrict__ quant,
                             float* __restrict__ idx_out)
{
    const int lane  = threadIdx.x & 31;
    const int wave  = threadIdx.x >> 5;
    const int qblk  = blockIdx.x * 8 + wave;     // 0..127
    const int qbase = qblk * 16;
    const int m     = lane & 15;                 // row/col within 16-half
    const int koff  = (lane >> 4) << 1;          // lanes 0-15: K%4 in {0,1}; 16-31: {2,3}

    // A fragments (queries): 32 k-steps of K=4, held in registers for the sweep.
    v2f a[32];
    const float* ap = slots + (size_t)(qbase + m) * H + koff;
#pragma unroll
    for (int kk = 0; kk < 32; ++kk) {
        a[kk].x = ap[4 * kk + 0];
        a[kk].y = ap[4 * kk + 1];
    }

    float bestv[8];
    int   besti[8];
#pragma unroll
    for (int r = 0; r < 8; ++r) { bestv[r] = 3.4e38f; besti[r] = 0; }

    for (int pt = 0; pt < P / 16; ++pt) {
        const int pbase = pt * 16;
        v8f c = {0.f, 0.f, 0.f, 0.f, 0.f, 0.f, 0.f, 0.f};
        float pn = 0.f;
        const float* bp = pool + (size_t)(pbase + m) * H + koff;
#pragma unroll
        for (int kk = 0; kk < 32; ++kk) {
            v2f b;
            b.x = bp[4 * kk + 0];
            b.y = bp[4 * kk + 1];
            pn += b.x * b.x + b.y * b.y;     // pool row norm, half of it per lane-half
            // 8 args: (neg_a, A, neg_b, B, c_mod, C, reuse_a, reuse_b)
            c = __builtin_amdgcn_wmma_f32_16x16x4_f32(
                    false, a[kk], false, b, (short)0, c, false, false);
        }
        pn += __shfl_xor(pn, 16);            // combine complementary K halves -> ||p||^2

#pragma unroll
        for (int r = 0; r < 8; ++r) {
            // C layout: VGPR r holds rows {r (lanes 0-15), r+8 (lanes 16-31)}, N = lane%16
            float sc = pn - 2.f * c[r];
            int   ci = pbase + m;
#pragma unroll
            for (int d = 8; d >= 1; d >>= 1) {   // min-reduce within each 16-lane half
                float ov = __shfl_xor(sc, d);
                int   oi = __shfl_xor(ci, d);
                if (ov < sc || (ov == sc && oi < ci)) { sc = ov; ci = oi; }
            }
            if (sc < bestv[r]) { bestv[r] = sc; besti[r] = ci; }
        }
    }

    // lanes 0 / 16 hold results for rows 0..7 / 8..15
    if (lane == 0 || lane == 16) {
        const int rowoff = (lane == 16) ? 8 : 0;
#pragma unroll
        for (int r = 0; r < 8; ++r)
            idx_out[qbase + rowoff + r] = (float)besti[r];
    }

    // quant = pool[idx] : whole wave copies each 128-float row (1 float4/lane)
#pragma unroll
    for (int mm = 0; mm < 16; ++mm) {
        int id = __shfl(besti[mm & 7], (mm < 8) ? 0 : 16);
        const float4* src = (const float4*)(pool  + (size_t)id * H);
        float4*       dst = (float4*)(quant + (size_t)(qbase + mm) * H);
        dst[lane] = src[lane];
    }
}

// ---------------------------------------------------------------------------
// Kernel 2: order-dependent EMA pool maintenance (sequential scan).
// Single persistent workgroup: each step does a full-pool argmin (block
// reduction in LDS) against the *current* pool, then EMA-updates one row.
// ---------------------------------------------------------------------------
__global__ __launch_bounds__(1024)
void slotprompt_maintain_kernel(const float* __restrict__ slots,
                                float* __restrict__ np)   // working/new pool
{
    __shared__ float s[H];
    __shared__ float rv[1024];
    __shared__ int   ri[1024];
    const int tid = threadIdx.x;

    for (int n = 0; n < NFLAT; ++n) {
        if (tid < H) s[tid] = slots[(size_t)n * H + tid];
        __syncthreads();

        float bv = 3.4e38f;
        int   bi = 0;
        const float4* sp = (const float4*)s;
        for (int k = 0; k < P / 1024; ++k) {
            const int p = k * 1024 + tid;
            const float4* rp = (const float4*)(np + (size_t)p * H);
            float acc = 0.f;
#pragma unroll 8
            for (int q = 0; q < H / 4; ++q) {
                float4 v = rp[q], w = sp[q];
                float dx = v.x - w.x, dy = v.y - w.y;
                float dz = v.z - w.z, dw = v.w - w.w;
                acc += dx * dx + dy * dy + dz * dz + dw * dw;
            }
            if (acc < bv) { bv = acc; bi = p; }   // p ascending -> first-index ties
        }

        rv[tid] = bv; ri[tid] = bi;
        __syncthreads();
        for (int off = 512; off > 0; off >>= 1) {
            if (tid < off) {
                float ov = rv[tid + off];
                int   oi = ri[tid + off];
                if (ov < rv[tid] || (ov == rv[tid] && oi < ri[tid])) {
                    rv[tid] = ov; ri[tid] = oi;
                }
            }
            __syncthreads();
        }

        const int imin = ri[0];
        if (tid < H) {
            float* row = np + (size_t)imin * H;
            row[tid] = ALPHA * row[tid] + (1.f - ALPHA) * s[tid];
        }
        __threadfence_block();
        __syncthreads();
    }
}

// ---------------------------------------------------------------------------
extern "C" void kernel_launch(void* const* d_in, const int* in_sizes, int n_in,
                              void* d_out, int out_size, void* d_ws, size_t ws_size,
                              hipStream_t stream)
{
    (void)in_sizes; (void)n_in; (void)out_size; (void)d_ws; (void)ws_size;

    const float* slots = (const float*)d_in[0];   // [256,8,128] fp32
    const float* pool  = (const float*)d_in[1];   // [16384,128] fp32

    float* out   = (float*)d_out;
    float* quant = out;                               // 2048*128
    float* idxf  = out + (size_t)NFLAT * H;           // 2048
    float* npool = idxf + NFLAT;                      // 16384*128 (working copy)

    // Working copy of pool lives in the new_pool output region (inputs immutable).
    hipMemcpyAsync(npool, pool, (size_t)P * H * sizeof(float),
                   hipMemcpyDeviceToDevice, stream);

    slotprompt_match_kernel<<<16, 256, 0, stream>>>(slots, pool, quant, idxf);
    slotprompt_maintain_kernel<<<1, 1024, 0, stream>>>(slots, npool);
}